// FactorGraphNeuralNetwork_48859547959500
// MI455X (gfx1250) — compile-verified
//
#include <hip/hip_runtime.h>
#include <hip/hip_bf16.h>

// FactorGraphNeuralNetwork on MI455X (gfx1250, wave32).
// All matmuls via V_WMMA_F32_16X16X4_F32 (full fp32 precision, matching the
// fp32 reference). H (532480 x 64 f32 = 136MB) lives in d_ws and fits in the
// 192MB L2. Problem is ~0.3 TFLOP total -> elementwise/latency bound, so fp32
// WMMA is the right precision/path. LN lane reductions use DS_SWIZZLE xor
// mode; the H tile fill uses GLOBAL_LOAD_ASYNC_TO_LDS_B128 when available.

#define BB 32
#define MM 256
#define KK 65
#define DD 64
#define NTOK (BB * MM * KK)       // 532480
#define NTILE (NTOK / 16)         // 33280
#define WAVES_PER_BLOCK 4
#define LDSTRIDE 68               // floats; 272B row, 16B aligned, conflict-spread

typedef __attribute__((ext_vector_type(2))) float v2f;
typedef __attribute__((ext_vector_type(8))) float v8f;
typedef int v4i_vs __attribute__((vector_size(16)));  // matches async-lds builtin param

// ---------------------------------------------------------------------------
// helpers
// ---------------------------------------------------------------------------

__device__ __forceinline__ float sigf(float x) { return 1.0f / (1.0f + __expf(-x)); }

// xor-exchange within groups of 32 lanes via DS_SWIZZLE (group-of-32 mode:
// imm = xor_mask<<10 | and_mask 0x1f). Masks < 16 stay within 16-lane halves.
#if __has_builtin(__builtin_amdgcn_ds_swizzle)
template <int IMM>
__device__ __forceinline__ float swz_add(float x) {
    return x + __int_as_float(__builtin_amdgcn_ds_swizzle(__float_as_int(x), IMM));
}
#else
template <int IMM>
__device__ __forceinline__ float swz_add(float x) {
    return x + __shfl_xor(x, (IMM >> 10), 16);
}
#endif
__device__ __forceinline__ float red16_sum(float x) {
    x = swz_add<0x041F>(x);   // xor 1
    x = swz_add<0x081F>(x);   // xor 2
    x = swz_add<0x101F>(x);   // xor 4
    x = swz_add<0x201F>(x);   // xor 8
    return x;
}

// Broadcast bias b[nb + (lane%16)] into all 8 accumulator rows (C/D layout:
// column n = lane%16 is constant across the 8 VGPRs).
__device__ __forceinline__ v8f bias8(const float* __restrict__ b, int nb) {
    int nl = threadIdx.x & 15;
    float x = b[nb + nl];
    v8f r;
#pragma unroll
    for (int i = 0; i < 8; ++i) r[i] = x;
    return r;
}

// acc += X(16x64, rows in aIn with stride LDSTRIDE) @ W[nrow:nrow+16, 0:64]^T
// via 16 chained V_WMMA_F32_16X16X4_F32.
// A layout (16x4 f32): lane L -> m=L%16, vgpr v -> k = 2*(L/16)+v  => b64 load.
// B layout (4x16 f32): lane L -> n=L%16, vgpr v -> k = 2*(L/16)+v  => b64 load
//   of W[n][k], W row-major [out][in] (we need B[k][n] = W^T).
__device__ __forceinline__ v8f wmma_k64(const float* aIn, const float* __restrict__ W,
                                        int nrow, v8f acc) {
    int lane = threadIdx.x & 31;
    int mn = lane & 15;
    int ks = (lane >> 4) << 1;
    const float* ap = aIn + mn * LDSTRIDE + ks;
    const float* bp = W + (nrow + mn) * DD + ks;
#pragma unroll
    for (int c = 0; c < 16; ++c) {
        v2f a = *(const v2f*)(ap + 4 * c);
        v2f b = *(const v2f*)(bp + 4 * c);
        acc = __builtin_amdgcn_wmma_f32_16x16x4_f32(false, a, false, b, (short)0, acc,
                                                    false, false);
    }
    return acc;
}

// Store one 16x16 C/D tile (tile index j -> dims 16j..16j+15) into a
// [16][LDSTRIDE] LDS region.
__device__ __forceinline__ void store_tile(v8f t, float* out, int j) {
    int lane = threadIdx.x & 31;
    int nl = lane & 15, hi = lane >> 4;
#pragma unroll
    for (int v = 0; v < 8; ++v) out[(v + 8 * hi) * LDSTRIDE + 16 * j + nl] = t[v];
}

// LayerNorm(64) + ReLU over 4 C/D tiles (one 16-token strip), result to LDS.
// Per token m = v + 8*(lane/16): its 64 values live in vgpr v across the 16
// lanes of one half-wave (4 tiles x 16 lanes) -> ds_swizzle xor reduction.
__device__ __forceinline__ void ln_relu_store(v8f* t, const float* __restrict__ g,
                                              const float* __restrict__ be, float* out) {
    int lane = threadIdx.x & 31;
    int nl = lane & 15, hi = lane >> 4;
    float gv[4], bv[4];
#pragma unroll
    for (int j = 0; j < 4; ++j) { gv[j] = g[16 * j + nl]; bv[j] = be[16 * j + nl]; }
#pragma unroll
    for (int v = 0; v < 8; ++v) {
        float s = 0.f, q = 0.f;
#pragma unroll
        for (int j = 0; j < 4; ++j) { float x = t[j][v]; s += x; q += x * x; }
        s = red16_sum(s);
        q = red16_sum(q);
        float mu = s * (1.0f / 64.0f);
        float var = q * (1.0f / 64.0f) - mu * mu;
        float inv = rsqrtf(var + 1e-5f);
#pragma unroll
        for (int j = 0; j < 4; ++j) {
            float y = (t[j][v] - mu) * inv * gv[j] + bv[j];
            out[(v + 8 * hi) * LDSTRIDE + 16 * j + nl] = fmaxf(y, 0.f);
        }
    }
}

// ---------------------------------------------------------------------------
// encoder: H = (relu(LN(x@w1^T+b1)))@w2^T + b2     x: [NTOK][4] -> H: [NTOK][64]
// ---------------------------------------------------------------------------
__global__ __launch_bounds__(128) void encoder_kernel(
    const float* __restrict__ x, float* __restrict__ H,
    const float* __restrict__ w1, const float* __restrict__ b1,
    const float* __restrict__ g, const float* __restrict__ be,
    const float* __restrict__ w2, const float* __restrict__ b2) {
    __shared__ float lds[WAVES_PER_BLOCK * 16 * LDSTRIDE];
    int wave = threadIdx.x >> 5, lane = threadIdx.x & 31;
    float* mid = lds + wave * 16 * LDSTRIDE;
    long tb = ((long)blockIdx.x * WAVES_PER_BLOCK + wave) * 16;
    int nl = lane & 15, hi = lane >> 4, ks = hi << 1;

    // A = x tile (16x4): single K=4 chunk, straight from global.
    v2f a = *(const v2f*)(x + (tb + nl) * 4 + ks);
    v8f t[4];
#pragma unroll
    for (int j = 0; j < 4; ++j) {
        v2f b = *(const v2f*)(w1 + (16 * j + nl) * 4 + ks);  // B[k][n] = w1[n][k]
        t[j] = __builtin_amdgcn_wmma_f32_16x16x4_f32(false, a, false, b, (short)0,
                                                     bias8(b1, 16 * j), false, false);
    }
    ln_relu_store(t, g, be, mid);
#pragma unroll
    for (int j = 0; j < 4; ++j) {
        v8f o = wmma_k64(mid, w2, 16 * j, bias8(b2, 16 * j));
#pragma unroll
        for (int v = 0; v < 8; ++v)
            H[(tb + v + 8 * hi) * DD + 16 * j + nl] = o[v];
    }
}

// ---------------------------------------------------------------------------
// max reductions
// ---------------------------------------------------------------------------
__global__ __launch_bounds__(64) void rowmax_kernel(const float* __restrict__ H,
                                                    float* __restrict__ red) {
    // one block per (b,m) row; thread d reduces over K1=65 tokens
    long r = blockIdx.x;
    int d = threadIdx.x;
    const float* p = H + r * KK * DD + d;
    float mx = -3.402823466e38f;
    for (int k = 0; k < KK; ++k) mx = fmaxf(mx, p[k * DD]);
    red[r * DD + d] = mx;
}

__global__ __launch_bounds__(64) void colmax_kernel(const float* __restrict__ H,
                                                    float* __restrict__ red) {
    // one block per (b,k1); thread d reduces over M=256 tokens
    int r = blockIdx.x;
    int d = threadIdx.x;
    int b = r / KK, k1 = r % KK;
    const float* p = H + (((long)b * MM) * KK + k1) * DD + d;
    float mx = -3.402823466e38f;
    for (int k = 0; k < MM; ++k) mx = fmaxf(mx, p[(long)k * KK * DD]);
    red[(long)r * DD + d] = mx;
}

// ---------------------------------------------------------------------------
// fused: T = MLP(H - max);  H = GRU(T, H)
// mode 0: max per row (rid = t / 65); mode 1: max per col (rid = b*65 + k1)
// ---------------------------------------------------------------------------
__global__ __launch_bounds__(128) void fused_gnn_kernel(
    float* __restrict__ H, const float* __restrict__ red, int mode,
    const float* __restrict__ w1, const float* __restrict__ b1,
    const float* __restrict__ g, const float* __restrict__ be,
    const float* __restrict__ w2, const float* __restrict__ b2,
    const float* __restrict__ wih, const float* __restrict__ bih,
    const float* __restrict__ whh, const float* __restrict__ bhh) {
    __shared__ float lds[WAVES_PER_BLOCK * 3 * 16 * LDSTRIDE];
    int wave = threadIdx.x >> 5, lane = threadIdx.x & 31;
    float* lH = lds + wave * 3 * 16 * LDSTRIDE;  // old hidden state
    float* lX = lH + 16 * LDSTRIDE;              // H - max, later reused for T
    float* lM = lX + 16 * LDSTRIDE;              // LN/ReLU mid activations
    long tb = ((long)blockIdx.x * WAVES_PER_BLOCK + wave) * 16;
    int nl = lane & 15, hi = lane >> 4;

    // Fill lH: 16 tokens x 64 f32 = 256 float4s; per-wave private slice.
#if __has_builtin(__builtin_amdgcn_global_load_async_to_lds_b128)
#pragma unroll
    for (int i = 0; i < 8; ++i) {
        int e = i * 32 + lane;
        int tok = e >> 4, d4 = e & 15;
        __builtin_amdgcn_global_load_async_to_lds_b128(
            (__attribute__((address_space(1))) v4i_vs*)(H + (tb + tok) * DD + d4 * 4),
            (__attribute__((address_space(3))) v4i_vs*)(lH + tok * LDSTRIDE + d4 * 4),
            0, 0);
    }
    asm volatile("s_wait_asynccnt 0x0" ::: "memory");
#else
#pragma unroll
    for (int i = 0; i < 8; ++i) {
        int e = i * 32 + lane;
        int tok = e >> 4, d4 = e & 15;
        float4 v = *(const float4*)(H + (tb + tok) * DD + d4 * 4);
        *(float4*)(lH + tok * LDSTRIDE + d4 * 4) = v;
    }
#endif

    // lX = lH - broadcast max (row or col); t is wave-uniform per tok, so the
    // /65, %65 become SALU and red row base becomes an SGPR pointer.
#pragma unroll
    for (int tok = 0; tok < 16; ++tok) {
        long t = tb + tok;
        long rid = mode ? ((t / ((long)MM * KK)) * KK + (t % KK)) : (t / KK);
        const float* rp = red + rid * DD;
        lX[tok * LDSTRIDE + lane] = lH[tok * LDSTRIDE + lane] - rp[lane];
        lX[tok * LDSTRIDE + 32 + lane] = lH[tok * LDSTRIDE + 32 + lane] - rp[32 + lane];
    }

    // ---- MLP: h1 = relu(LN(lX@w1^T+b1)); T = h1@w2^T+b2 -> stored into lX
    v8f h1[4];
#pragma unroll
    for (int j = 0; j < 4; ++j) h1[j] = wmma_k64(lX, w1, 16 * j, bias8(b1, 16 * j));
    ln_relu_store(h1, g, be, lM);
#pragma unroll
    for (int j = 0; j < 4; ++j)
        store_tile(wmma_k64(lM, w2, 16 * j, bias8(b2, 16 * j)), lX, j);

    // ---- GRU: gi = T@wih^T+bih, gh = H@whh^T+bhh; gates r,z,n (PyTorch order)
    v8f r[4], z[4];
#pragma unroll
    for (int j = 0; j < 4; ++j) {
        v8f gi = wmma_k64(lX, wih, 16 * j, bias8(bih, 16 * j));
        v8f gh = wmma_k64(lH, whh, 16 * j, bias8(bhh, 16 * j));
#pragma unroll
        for (int v = 0; v < 8; ++v) r[j][v] = sigf(gi[v] + gh[v]);
    }
#pragma unroll
    for (int j = 0; j < 4; ++j) {
        v8f gi = wmma_k64(lX, wih, 64 + 16 * j, bias8(bih, 64 + 16 * j));
        v8f gh = wmma_k64(lH, whh, 64 + 16 * j, bias8(bhh, 64 + 16 * j));
#pragma unroll
        for (int v = 0; v < 8; ++v) z[j][v] = sigf(gi[v] + gh[v]);
    }
#pragma unroll
    for (int j = 0; j < 4; ++j) {
        v8f gi = wmma_k64(lX, wih, 128 + 16 * j, bias8(bih, 128 + 16 * j));
        v8f gh = wmma_k64(lH, whh, 128 + 16 * j, bias8(bhh, 128 + 16 * j));
#pragma unroll
        for (int v = 0; v < 8; ++v) {
            float nn = tanhf(gi[v] + r[j][v] * gh[v]);
            float hold = lH[(v + 8 * hi) * LDSTRIDE + 16 * j + nl];
            float zz = z[j][v];
            H[(tb + v + 8 * hi) * DD + 16 * j + nl] = (1.0f - zz) * nn + zz * hold;
        }
    }
}

// ---------------------------------------------------------------------------
// head: logits[t] = dot(H[t], head_w) + head_b
// ---------------------------------------------------------------------------
__global__ __launch_bounds__(256) void head_kernel(const float* __restrict__ H,
                                                   const float* __restrict__ hw,
                                                   const float* __restrict__ hb,
                                                   float* __restrict__ out) {
    long t = (long)blockIdx.x * blockDim.x + threadIdx.x;
    if (t >= NTOK) return;
    const float4* h4 = (const float4*)(H + t * DD);
    const float4* w4 = (const float4*)hw;
    float acc = 0.f;
#pragma unroll
    for (int i = 0; i < 16; ++i) {
        float4 a = h4[i], w = w4[i];
        acc += a.x * w.x + a.y * w.y + a.z * w.z + a.w * w.w;
    }
    out[t] = acc + hb[0];
}

// ---------------------------------------------------------------------------
// launch
// ---------------------------------------------------------------------------
// d_in flattening (setup_inputs dict order, depth-first):
//  0: hybrid_input
//  1..6:  encoder {w1,b1,g,be,w2,b2}
//  7+L*20 + {0..5}:   layers[L].mlp_row {w1,b1,g,be,w2,b2}
//  7+L*20 + {6..11}:  layers[L].mlp_col {w1,b1,g,be,w2,b2}
//  7+L*20 + {12..15}: layers[L].gru_row {wih,bih,whh,bhh}
//  7+L*20 + {16..19}: layers[L].gru_col {wih,bih,whh,bhh}
//  47: head_w   48: head_b
extern "C" void kernel_launch(void* const* d_in, const int* in_sizes, int n_in,
                              void* d_out, int out_size, void* d_ws, size_t ws_size,
                              hipStream_t stream) {
    auto f = [&](int i) { return (const float*)d_in[i]; };
    const float* x = f(0);
    float* H = (float*)d_ws;                       // NTOK*64 floats (136 MB)
    float* red = H + (size_t)NTOK * DD;            // max(8192,2080)*64 floats

    const int nblk = NTILE / WAVES_PER_BLOCK;      // 8320

    encoder_kernel<<<nblk, 128, 0, stream>>>(x, H, f(1), f(2), f(3), f(4), f(5), f(6));

    for (int L = 0; L < 2; ++L) {
        int b = 7 + L * 20;
        rowmax_kernel<<<BB * MM, 64, 0, stream>>>(H, red);
        fused_gnn_kernel<<<nblk, 128, 0, stream>>>(
            H, red, 0, f(b + 0), f(b + 1), f(b + 2), f(b + 3), f(b + 4), f(b + 5),
            f(b + 12), f(b + 13), f(b + 14), f(b + 15));
        colmax_kernel<<<BB * KK, 64, 0, stream>>>(H, red);
        fused_gnn_kernel<<<nblk, 128, 0, stream>>>(
            H, red, 1, f(b + 6), f(b + 7), f(b + 8), f(b + 9), f(b + 10), f(b + 11),
            f(b + 16), f(b + 17), f(b + 18), f(b + 19));
    }

    head_kernel<<<NTOK / 256, 256, 0, stream>>>(H, f(47), f(48), (float*)d_out);
}